// OptNet_24902220382768
// MI455X (gfx1250) — compile-verified
//
#include <hip/hip_runtime.h>
#include <math.h>

typedef __attribute__((ext_vector_type(2))) float v2f;
typedef __attribute__((ext_vector_type(8))) float v8f;

#define DD     128
#define NITERS 25
#define SIGMA  0.1f
#define GS     129   // G stride      (odd  -> conflict-free rows & cols)
#define MS     130   // M stride      (even -> 8B-aligned k-pairs for WMMA frags)
#define BTS    132   // Bt stride     (even, lanes hit disjoint bank pairs)
#define NT     256
#define NB     16

__global__ __launch_bounds__(NT) void optnet_ipm_kernel(
    const float* __restrict__ G, const float* __restrict__ h,
    const float* __restrict__ x, float* __restrict__ out)
{
    extern __shared__ float smem[];
    float* Gs  = smem;             // 128*129 : G row-major
    float* Msh = Gs  + DD*GS;      // 128*130 : KKT matrix -> LU factors in place
    float* Bt  = Msh + DD*MS;      // 128*132 : Bt[c][k] = sqrt(w[k])*G[k][c]
    float* xs  = Bt  + DD*BTS;
    float* hs  = xs  + DD;
    float* zv  = hs  + DD;
    float* sv  = zv  + DD;
    float* lv  = sv  + DD;
    float* Gz  = lv  + DD;
    float* rpr = Gz  + DD;
    float* wv  = rpr + DD;
    float* sws = wv  + DD;
    float* rhm = sws + DD;
    float* rv  = rhm + DD;         // rhs -> dz
    float* red  = rv  + DD;        // 256
    float* red2 = red + NT;        // 256

    const int t      = threadIdx.x;
    const int b      = blockIdx.x;
    const int lane   = t & 31;
    const int wave   = t >> 5;
    const int laneM  = lane & 15;
    const int koff   = (lane >> 4) << 1;  // frag K sub-offset: 0 / 2
    const int rowoff = (lane >> 4) << 3;  // C/D frag: M = r / r+8

    for (int idx = t; idx < DD*DD; idx += NT)
        Gs[(idx >> 7)*GS + (idx & 127)] = G[idx];
    if (t < DD) {
        hs[t] = h[t];
        xs[t] = x[b*DD + t];
        zv[t] = 0.0f;
        sv[t] = fmaxf(hs[t], 1.0f);
        lv[t] = 1.0f;
    }
    __syncthreads();

    const int hrow = t & 127;
    const int half = t >> 7;
    const int j0h  = half * 64;

    for (int it = 0; it < NITERS; ++it) {
        // ================= Gz = G z (row GEMV, 2 threads/row) =================
        {
            float acc = 0.f;
            const float* gr = &Gs[hrow*GS + j0h];
            #pragma unroll 8
            for (int j = 0; j < 64; ++j) acc += gr[j] * zv[j0h + j];
            red[t] = acc;
        }
        __syncthreads();
        if (t < DD) Gz[t] = red[t] + red[t + 128];
        __syncthreads();

        // ================= residuals, mu =================
        if (t < DD) { rpr[t] = Gz[t] + sv[t] - hs[t]; red[t] = lv[t]*sv[t]; }
        else        red[t] = 0.f;
        __syncthreads();
        for (int off = 128; off > 0; off >>= 1) {
            if (t < off) red[t] += red[t + off];
            __syncthreads();
        }
        float mu = red[0] * (1.0f / 128.0f);
        if (t < DD) {
            float rc = lv[t]*sv[t] - SIGMA*mu;
            float wk = lv[t]/sv[t];
            wv[t]  = wk;
            sws[t] = sqrtf(wk);
            rhm[t] = (-rc + lv[t]*rpr[t]) / sv[t];
        }
        __syncthreads();

        // ======= build scaled transposed Bt (for b64 WMMA frags) =======
        for (int idx = t; idx < DD*DD; idx += NT) {
            int k = idx & 127, c = idx >> 7;
            Bt[c*BTS + k] = sws[k] * Gs[k*GS + c];
        }
        // ======= rv = -r_dual - rhs_m @ G (two column GEMVs fused) =======
        {
            float a1 = 0.f, a2 = 0.f;
            #pragma unroll 8
            for (int k = 0; k < 64; ++k) {
                int kk = j0h + k;
                float g = Gs[kk*GS + hrow];
                a1 += lv[kk]  * g;
                a2 += rhm[kk] * g;
            }
            red[t] = a1; red2[t] = a2;
        }
        __syncthreads();
        if (t < DD) {
            float lamG = red[t]  + red[t+128];
            float rmG  = red2[t] + red2[t+128];
            rv[t] = -(2.0f*(zv[t]-xs[t]) + lamG) - rmG;
        }
        __syncthreads();

        // ========== M = 2I + G^T diag(w) G : WMMA, upper tiles only ==========
        {
            const int ntiles = (wave < 4) ? 5 : 4;      // 36 tiles over 8 waves
            for (int tn = 0; tn < ntiles; ++tn) {
                int tt = wave + 8*tn;
                int rem = tt, ti = 0;
                while (rem >= 8 - ti) { rem -= 8 - ti; ++ti; }
                const int tj = ti + rem;                // ti <= tj
                const int i0 = ti*16, j0 = tj*16;
                const float* ap = &Bt[(i0 + laneM)*BTS + koff];
                const float* bp = &Bt[(j0 + laneM)*BTS + koff];
                v8f c = {};
                for (int g = 0; g < 4; ++g) {           // 4 groups of 8 k-steps
                    v2f af[8], bf[8];
                    #pragma unroll
                    for (int u = 0; u < 8; ++u) {
                        af[u] = *(const v2f*)(ap + g*32 + u*4);
                        bf[u] = *(const v2f*)(bp + g*32 + u*4);
                    }
                    #pragma unroll
                    for (int u = 0; u < 8; ++u)
                        c = __builtin_amdgcn_wmma_f32_16x16x4_f32(
                                false, af[u], false, bf[u], (short)0, c, false, false);
                }
                #pragma unroll
                for (int r = 0; r < 8; ++r) {
                    int row = i0 + rowoff + r;
                    int col = j0 + laneM;
                    float v = c[r];
                    if (row == col) v += 2.0f;
                    Msh[row*MS + col] = v;
                    if (ti != tj) Msh[col*MS + row] = v;   // mirror
                }
            }
        }
        __syncthreads();

        // ========== blocked LU (SPD, no pivot), rank-16 updates on WMMA ==========
        for (int kb = 0; kb < 8; ++kb) {
            const int k0 = kb*NB;
            // --- panel factorization (cols k0..k0+15, rows k0..127) ---
            for (int kk = 0; kk < NB; ++kk) {
                const int k = k0 + kk;
                const float piv = Msh[k*MS + k];
                const int i = k + 1 + t;
                if (i < DD) {
                    float f = Msh[i*MS + k] / piv;
                    Msh[i*MS + k] = f;
                    for (int j = kk + 1; j < NB; ++j)
                        Msh[i*MS + k0 + j] -= f * Msh[k*MS + k0 + j];
                }
                __syncthreads();
            }
            const int c0 = k0 + NB;
            const int W  = DD - c0;
            if (W > 0) {
                // --- U12 = L11^{-1} A12 : per-column register triangular solve ---
                if (t < W) {
                    const int c = c0 + t;
                    float a[NB];
                    #pragma unroll
                    for (int m = 0; m < NB; ++m) a[m] = Msh[(k0+m)*MS + c];
                    #pragma unroll
                    for (int kk = 1; kk < NB; ++kk)
                        #pragma unroll
                        for (int m = 0; m < kk; ++m)
                            a[kk] -= Msh[(k0+kk)*MS + k0+m] * a[m];
                    #pragma unroll
                    for (int m = 1; m < NB; ++m) Msh[(k0+m)*MS + c] = a[m];
                }
                __syncthreads();
                // --- trailing A22 -= L21 * U12 : WMMA rank-16 update ---
                const int nb = W >> 4;
                int idx = 0;
                for (int ti = 0; ti < nb; ++ti)
                    for (int tj = 0; tj < nb; ++tj, ++idx) {
                        if ((idx & 7) != wave) continue;
                        const int r0  = c0 + ti*16;
                        const int cc0 = c0 + tj*16;
                        v8f c = {};
                        #pragma unroll
                        for (int r = 0; r < 8; ++r)
                            c[r] = Msh[(r0 + rowoff + r)*MS + cc0 + laneM];
                        #pragma unroll
                        for (int ks = 0; ks < 4; ++ks) {
                            v2f af = *(const v2f*)&Msh[(r0 + laneM)*MS + k0 + ks*4 + koff];
                            af.x = -af.x; af.y = -af.y;     // f32 WMMA has no A-neg
                            v2f bf;
                            bf.x = Msh[(k0 + ks*4 + koff    )*MS + cc0 + laneM];
                            bf.y = Msh[(k0 + ks*4 + koff + 1)*MS + cc0 + laneM];
                            c = __builtin_amdgcn_wmma_f32_16x16x4_f32(
                                    false, af, false, bf, (short)0, c, false, false);
                        }
                        #pragma unroll
                        for (int r = 0; r < 8; ++r)
                            Msh[(r0 + rowoff + r)*MS + cc0 + laneM] = c[r];
                    }
            }
            __syncthreads();
        }

        // ========== blocked forward solve L y = rhs ==========
        for (int kb = 0; kb < 8; ++kb) {
            const int k0 = kb*NB;
            if (wave == 0) {                 // 16x16 micro-solve, shuffle broadcast
                const int l = lane & 15;
                float val = rv[k0 + l];
                #pragma unroll
                for (int kk = 0; kk < NB; ++kk) {
                    float bc = __shfl(val, kk, 32);
                    if (l > kk) val -= Msh[(k0 + l)*MS + k0 + kk] * bc;
                }
                if (lane < NB) rv[k0 + lane] = val;
            }
            __syncthreads();
            const int i = k0 + NB + t;       // rank-16 GEMV on remaining rows
            if (i < DD) {
                float acc = rv[i];
                #pragma unroll
                for (int kk = 0; kk < NB; ++kk)
                    acc -= Msh[i*MS + k0 + kk] * rv[k0 + kk];
                rv[i] = acc;
            }
            __syncthreads();
        }
        // ========== blocked backward solve U dz = y ==========
        for (int kb = 7; kb >= 0; --kb) {
            const int k0 = kb*NB;
            if (wave == 0) {
                const int l = lane & 15;
                float val = rv[k0 + l];
                #pragma unroll
                for (int kk = NB-1; kk >= 0; --kk) {
                    float d = Msh[(k0+kk)*MS + k0+kk];
                    if (l == kk) val /= d;
                    float bc = __shfl(val, kk, 32);
                    if (l < kk) val -= Msh[(k0 + l)*MS + k0 + kk] * bc;
                }
                if (lane < NB) rv[k0 + lane] = val;
            }
            __syncthreads();
            if (t < k0) {
                float acc = rv[t];
                #pragma unroll
                for (int kk = 0; kk < NB; ++kk)
                    acc -= Msh[t*MS + k0 + kk] * rv[k0 + kk];
                rv[t] = acc;
            }
            __syncthreads();
        }

        // ================= dGz = G dz (row GEMV) =================
        {
            float acc = 0.f;
            const float* gr = &Gs[hrow*GS + j0h];
            #pragma unroll 8
            for (int j = 0; j < 64; ++j) acc += gr[j] * rv[j0h + j];
            red[t] = acc;
        }
        __syncthreads();

        // ================= ds, dlam, alpha, update =================
        float ds_t = 0.f, dl_t = 0.f;
        float cand = __builtin_inff();
        if (t < DD) {
            float dgz = red[t] + red[t+128];
            ds_t = -rpr[t] - dgz;
            dl_t = rhm[t] + wv[t]*dgz;
            float c1 = (ds_t < 0.f) ? (-sv[t]/ds_t) : __builtin_inff();
            float c2 = (dl_t < 0.f) ? (-lv[t]/dl_t) : __builtin_inff();
            cand = fminf(c1, c2);
        }
        __syncthreads();
        red[t] = cand;
        __syncthreads();
        for (int off = 128; off > 0; off >>= 1) {
            if (t < off) red[t] = fminf(red[t], red[t+off]);
            __syncthreads();
        }
        float alpha = fminf(1.0f, 0.99f * red[0]);
        if (t < DD) {
            zv[t] += alpha * rv[t];
            sv[t] += alpha * ds_t;
            lv[t] += alpha * dl_t;
        }
        __syncthreads();
    }

    if (t < DD) out[b*DD + t] = zv[t];
}

extern "C" void kernel_launch(void* const* d_in, const int* in_sizes, int n_in,
                              void* d_out, int out_size, void* d_ws, size_t ws_size,
                              hipStream_t stream) {
    const float* G = (const float*)d_in[0];   // (128,128)
    const float* h = (const float*)d_in[1];   // (128,)
    const float* x = (const float*)d_in[2];   // (B,128)
    float* out = (float*)d_out;
    const int B = in_sizes[2] / DD;           // 2048

    const size_t shmem = (size_t)(DD*GS + DD*MS + DD*BTS + 11*DD + 2*NT) * sizeof(float);
    hipFuncSetAttribute((const void*)optnet_ipm_kernel,
                        hipFuncAttributeMaxDynamicSharedMemorySize, (int)shmem);
    optnet_ipm_kernel<<<B, NT, shmem, stream>>>(G, h, x, out);
}